// DeepResNetMLP_51591147159760
// MI455X (gfx1250) — compile-verified
//
#include <hip/hip_runtime.h>

// ---------------------------------------------------------------------------
// DeepResNetMLP for MI455X (gfx1250, wave32).
// Memory-bound pipeline (≈8 GB traffic @ 23.3 TB/s -> ~350us floor); fp32
// WMMA 16x16x4 keeps full reference precision on the matrix pipe.
// BN(h) = s*h + t is folded into the NEXT kernel's activation read, so each
// layer is one streaming matmul pass + a tiny BN-affine finalize kernel.
// All elementwise streaming is float4 (b128) for full-line coalescing.
// ---------------------------------------------------------------------------

typedef __attribute__((ext_vector_type(2))) float v2f;
typedef __attribute__((ext_vector_type(4))) float v4f;
typedef __attribute__((ext_vector_type(8))) float v8f;

#define NROWS          (1u << 20)
#define DIN            64
#define HDIM           22
#define HP             32                     // padded feature dim
#define NGROUPS        1024
#define TPB            256
#define NWAVES         8
#define ROWS_PER_GROUP (NROWS / NGROUPS)      // 1024
#define ROWS_PER_WAVE  (ROWS_PER_GROUP / NWAVES) // 128
#define TILES_PER_WAVE (ROWS_PER_WAVE / 16)   // 8
#define KSTEPS_H       6                      // ceil(22/4)
#define KSTEPS_D       16                     // 64/4

__device__ __forceinline__ v8f wmma_f32(v2f a, v2f b, v8f c) {
  // D = A(16x4 f32) * B(4x16 f32) + C(16x16 f32)
  return __builtin_amdgcn_wmma_f32_16x16x4_f32(
      /*neg_a=*/false, a, /*neg_b=*/false, b,
      /*c_mod=*/(short)0, c, /*reuse_a=*/false, /*reuse_b=*/false);
}

__device__ __forceinline__ float leaky(float z) {
  return z >= 0.0f ? z : 0.01f * z;
}
__device__ __forceinline__ v4f leaky4(v4f z) {
  v4f r;
  r.x = leaky(z.x); r.y = leaky(z.y); r.z = leaky(z.z); r.w = leaky(z.w);
  return r;
}

// B-operand fetch: B[k][j] = W[j*kd + k], zero-padded outside (HDIM, kd)
__device__ __forceinline__ v2f loadB(const float* __restrict__ W,
                                     int j, int k, int kd) {
  v2f b;
  bool ok = (j < HDIM) & (k < kd);
  b.x = ok ? W[j * kd + k] : 0.0f;
  b.y = (ok && (k + 1 < kd)) ? W[j * kd + k + 1] : 0.0f;
  return b;
}

// ---------------------------------------------------------------------------
// Kernel 1: input projection  H = x @ Wp.T + bp   (+ per-feature stats)
// ---------------------------------------------------------------------------
__global__ __launch_bounds__(TPB) void proj_kernel(
    const float* __restrict__ x, const float* __restrict__ Wp,
    const float* __restrict__ bp, float* __restrict__ Hb,
    float* __restrict__ partials) {
  __shared__ float sSum[HP], sSsq[HP], sB[HP];
  const int tid = threadIdx.x, lane = tid & 31, wave = tid >> 5, g = blockIdx.x;
  if (tid < HP) {
    sSum[tid] = 0.0f; sSsq[tid] = 0.0f;
    sB[tid] = (tid < HDIM) ? bp[tid] : 0.0f;
  }
  __syncthreads();

  const int arow = lane & 15;         // A-matrix row within tile
  const int koff = (lane >> 4) * 2;   // K sub-offset per half-wave
  const int bcol = lane & 15;         // B/C column
  const int half = lane >> 4;         // C row half (0 / +8)

  // Preload B operands (weights) into registers: 2 N-tiles x 16 K-steps.
  v2f B0[KSTEPS_D], B1[KSTEPS_D];
#pragma unroll
  for (int ks = 0; ks < KSTEPS_D; ++ks) {
    const int k = ks * 4 + koff;
    B0[ks] = loadB(Wp, bcol, k, DIN);
    B1[ks] = loadB(Wp, bcol + 16, k, DIN);
  }

  float ls0 = 0, lq0 = 0, ls1 = 0, lq1 = 0;
  const int rbase = g * ROWS_PER_GROUP + wave * ROWS_PER_WAVE;
  for (int t = 0; t < TILES_PER_WAVE; ++t) {
    const int rb = rbase + t * 16;
    const float* xr = x + (size_t)(rb + arow) * DIN;
    __builtin_prefetch(xr + 16 * DIN, 0, 1);  // next tile -> global_prefetch_b8
    v8f acc0 = {}; v8f acc1 = {};
#pragma unroll
    for (int ks = 0; ks < KSTEPS_D; ++ks) {
      v2f a = *(const v2f*)(xr + ks * 4 + koff);
      acc0 = wmma_f32(a, B0[ks], acc0);
      acc1 = wmma_f32(a, B1[ks], acc1);
    }
    const float bb0 = sB[bcol], bb1 = sB[bcol + 16];
#pragma unroll
    for (int r = 0; r < 8; ++r) {
      const int row = rb + r + half * 8;
      float h0 = acc0[r] + bb0;
      float h1 = acc1[r] + bb1;
      Hb[(size_t)row * HP + bcol]      = h0;
      Hb[(size_t)row * HP + 16 + bcol] = h1;
      ls0 += h0; lq0 += h0 * h0;
      ls1 += h1; lq1 += h1 * h1;
    }
  }
  atomicAdd(&sSum[bcol], ls0);      atomicAdd(&sSsq[bcol], lq0);
  atomicAdd(&sSum[bcol + 16], ls1); atomicAdd(&sSsq[bcol + 16], lq1);
  __syncthreads();
  if (tid < HP) {
    partials[g * 64 + tid]      = sSum[tid];
    partials[g * 64 + 32 + tid] = sSsq[tid];
  }
}

// ---------------------------------------------------------------------------
// Stat finalize: (sum, sumsq) over groups -> BN affine (s, t).
// 8 waves strip-mine the 1024 group partials (128 serial iters each),
// then one wave folds + writes the affine. Cuts per-BN critical path ~8x.
// ---------------------------------------------------------------------------
__global__ __launch_bounds__(TPB) void stats_fin_kernel(
    const float* __restrict__ partials, const float* __restrict__ gamma,
    const float* __restrict__ beta, float* __restrict__ sc) {
  __shared__ float rs[NWAVES][HP], rq[NWAVES][HP];
  const int f = threadIdx.x & 31;
  const int chunk = threadIdx.x >> 5;  // 0..7
  float s = 0.0f, q = 0.0f;
  for (int i = chunk; i < NGROUPS; i += NWAVES) {
    s += partials[i * 64 + f];
    q += partials[i * 64 + 32 + f];
  }
  rs[chunk][f] = s; rq[chunk][f] = q;
  __syncthreads();
  if (threadIdx.x < HP) {
    s = 0.0f; q = 0.0f;
#pragma unroll
    for (int w = 0; w < NWAVES; ++w) { s += rs[w][f]; q += rq[w][f]; }
    const float invN = 1.0f / (float)NROWS;
    const float mu = s * invN;
    const float var = q * invN - mu * mu;
    float sf = 0.0f, tf = 0.0f;
    if (f < HDIM) {
      const float r = rsqrtf(var + 1e-5f);
      sf = gamma[f] * r;
      tf = beta[f] - mu * sf;
    }
    sc[f] = sf;
    sc[32 + f] = tf;
  }
}

// ---------------------------------------------------------------------------
// Block first half:  carry = mode ? leaky(C + s*H + t) : s*H + t
//                    H' = carry @ W1.T + b1   (+ stats of H')
// ---------------------------------------------------------------------------
__global__ __launch_bounds__(TPB) void block1_kernel(
    float* __restrict__ Hb, float* __restrict__ Cb,
    const float* __restrict__ scst, const float* __restrict__ W,
    const float* __restrict__ bvec, float* __restrict__ partials,
    const int mode) {
  __shared__ __align__(16) float At[NWAVES][16][HP];  // per-wave A tile (16 KB)
  __shared__ __align__(16) float sS[HP], sT[HP];
  __shared__ float sSum[HP], sSsq[HP], sB[HP];
  const int tid = threadIdx.x, lane = tid & 31, wave = tid >> 5, g = blockIdx.x;
  if (tid < HP) {
    sSum[tid] = 0.0f; sSsq[tid] = 0.0f;
    sB[tid] = (tid < HDIM) ? bvec[tid] : 0.0f;
    sS[tid] = scst[tid]; sT[tid] = scst[32 + tid];
  }
  __syncthreads();

  const int arow = lane & 15, koff = (lane >> 4) * 2;
  const int bcol = lane & 15, half = lane >> 4;

  v2f B0[KSTEPS_H], B1[KSTEPS_H];
#pragma unroll
  for (int ks = 0; ks < KSTEPS_H; ++ks) {
    const int k = ks * 4 + koff;
    B0[ks] = loadB(W, bcol, k, HDIM);
    B1[ks] = loadB(W, bcol + 16, k, HDIM);
  }

  float ls0 = 0, lq0 = 0, ls1 = 0, lq1 = 0;
  const int rbase = g * ROWS_PER_GROUP + wave * ROWS_PER_WAVE;
  for (int t = 0; t < TILES_PER_WAVE; ++t) {
    const int rb = rbase + t * 16;
    // elementwise (float4): BN-affine (+residual+leaky), write carry, stage A
#pragma unroll
    for (int qi = 0; qi < 4; ++qi) {
      const int flat = lane + 32 * qi;       // 0..127 float4 slots
      const int r = flat >> 3, c4 = (flat & 7) * 4;
      const size_t idx = (size_t)(rb + r) * HP + c4;
      v4f h = *(const v4f*)(Hb + idx);
      v4f s = *(const v4f*)&sS[c4];
      v4f tt = *(const v4f*)&sT[c4];
      v4f v = s * h + tt;
      if (mode) {
        v4f co = *(const v4f*)(Cb + idx);
        v = leaky4(co + v);
      }
      *(v4f*)(Cb + idx) = v;
      *(v4f*)&At[wave][r][c4] = v;
    }
    asm volatile("s_wait_dscnt 0" ::: "memory");  // wave-local LDS RAW fence

    v8f acc0 = {}; v8f acc1 = {};
#pragma unroll
    for (int ks = 0; ks < KSTEPS_H; ++ks) {
      v2f a = *(const v2f*)&At[wave][arow][ks * 4 + koff];
      acc0 = wmma_f32(a, B0[ks], acc0);
      acc1 = wmma_f32(a, B1[ks], acc1);
    }
    const float bb0 = sB[bcol], bb1 = sB[bcol + 16];
#pragma unroll
    for (int r = 0; r < 8; ++r) {
      const int row = rb + r + half * 8;
      float h0 = acc0[r] + bb0;
      float h1 = acc1[r] + bb1;
      Hb[(size_t)row * HP + bcol]      = h0;
      Hb[(size_t)row * HP + 16 + bcol] = h1;
      ls0 += h0; lq0 += h0 * h0;
      ls1 += h1; lq1 += h1 * h1;
    }
  }
  atomicAdd(&sSum[bcol], ls0);      atomicAdd(&sSsq[bcol], lq0);
  atomicAdd(&sSum[bcol + 16], ls1); atomicAdd(&sSsq[bcol + 16], lq1);
  __syncthreads();
  if (tid < HP) {
    partials[g * 64 + tid]      = sSum[tid];
    partials[g * 64 + 32 + tid] = sSsq[tid];
  }
}

// ---------------------------------------------------------------------------
// Block second half: a = leaky(s*H + t);  H = a @ W2.T + b2 (in-place, + stats)
// ---------------------------------------------------------------------------
__global__ __launch_bounds__(TPB) void block2_kernel(
    float* __restrict__ Hb, const float* __restrict__ scst,
    const float* __restrict__ W, const float* __restrict__ bvec,
    float* __restrict__ partials) {
  __shared__ __align__(16) float At[NWAVES][16][HP];
  __shared__ __align__(16) float sS[HP], sT[HP];
  __shared__ float sSum[HP], sSsq[HP], sB[HP];
  const int tid = threadIdx.x, lane = tid & 31, wave = tid >> 5, g = blockIdx.x;
  if (tid < HP) {
    sSum[tid] = 0.0f; sSsq[tid] = 0.0f;
    sB[tid] = (tid < HDIM) ? bvec[tid] : 0.0f;
    sS[tid] = scst[tid]; sT[tid] = scst[32 + tid];
  }
  __syncthreads();

  const int arow = lane & 15, koff = (lane >> 4) * 2;
  const int bcol = lane & 15, half = lane >> 4;

  v2f B0[KSTEPS_H], B1[KSTEPS_H];
#pragma unroll
  for (int ks = 0; ks < KSTEPS_H; ++ks) {
    const int k = ks * 4 + koff;
    B0[ks] = loadB(W, bcol, k, HDIM);
    B1[ks] = loadB(W, bcol + 16, k, HDIM);
  }

  float ls0 = 0, lq0 = 0, ls1 = 0, lq1 = 0;
  const int rbase = g * ROWS_PER_GROUP + wave * ROWS_PER_WAVE;
  for (int t = 0; t < TILES_PER_WAVE; ++t) {
    const int rb = rbase + t * 16;
#pragma unroll
    for (int qi = 0; qi < 4; ++qi) {
      const int flat = lane + 32 * qi;
      const int r = flat >> 3, c4 = (flat & 7) * 4;
      const size_t idx = (size_t)(rb + r) * HP + c4;
      v4f h = *(const v4f*)(Hb + idx);
      v4f s = *(const v4f*)&sS[c4];
      v4f tt = *(const v4f*)&sT[c4];
      *(v4f*)&At[wave][r][c4] = leaky4(s * h + tt);
    }
    asm volatile("s_wait_dscnt 0" ::: "memory");

    v8f acc0 = {}; v8f acc1 = {};
#pragma unroll
    for (int ks = 0; ks < KSTEPS_H; ++ks) {
      v2f a = *(const v2f*)&At[wave][arow][ks * 4 + koff];
      acc0 = wmma_f32(a, B0[ks], acc0);
      acc1 = wmma_f32(a, B1[ks], acc1);
    }
    const float bb0 = sB[bcol], bb1 = sB[bcol + 16];
#pragma unroll
    for (int r = 0; r < 8; ++r) {
      const int row = rb + r + half * 8;
      float h0 = acc0[r] + bb0;
      float h1 = acc1[r] + bb1;
      Hb[(size_t)row * HP + bcol]      = h0;  // in-place, row-local -> safe
      Hb[(size_t)row * HP + 16 + bcol] = h1;
      ls0 += h0; lq0 += h0 * h0;
      ls1 += h1; lq1 += h1 * h1;
    }
  }
  atomicAdd(&sSum[bcol], ls0);      atomicAdd(&sSsq[bcol], lq0);
  atomicAdd(&sSum[bcol + 16], ls1); atomicAdd(&sSsq[bcol + 16], lq1);
  __syncthreads();
  if (tid < HP) {
    partials[g * 64 + tid]      = sSum[tid];
    partials[g * 64 + 32 + tid] = sSsq[tid];
  }
}

// ---------------------------------------------------------------------------
// Final: hf = leaky(C + s*H + t);  out = hf @ Wo.T + bo   (O = 1)
// Padding cols 22..23 contribute 0 (s=t=0, carry padding=0, Wo padding=0).
// ---------------------------------------------------------------------------
__global__ __launch_bounds__(TPB) void final_kernel(
    const float* __restrict__ Hb, const float* __restrict__ Cb,
    const float* __restrict__ scst, const float* __restrict__ Wo,
    const float* __restrict__ bo, float* __restrict__ out) {
  __shared__ __align__(16) float sS[HP], sT[HP], sW[HP];
  const int tid = threadIdx.x;
  if (tid < HP) {
    sS[tid] = scst[tid]; sT[tid] = scst[32 + tid];
    sW[tid] = (tid < HDIM) ? Wo[tid] : 0.0f;
  }
  __syncthreads();
  const int row = blockIdx.x * TPB + tid;
  const float* hr = Hb + (size_t)row * HP;
  const float* cr = Cb + (size_t)row * HP;
  float acc = bo[0];
#pragma unroll
  for (int c4 = 0; c4 < 24; c4 += 4) {
    v4f h = *(const v4f*)(hr + c4);
    v4f cc = *(const v4f*)(cr + c4);
    v4f s = *(const v4f*)&sS[c4];
    v4f tt = *(const v4f*)&sT[c4];
    v4f w = *(const v4f*)&sW[c4];
    v4f v = leaky4(cc + s * h + tt);
    v4f p = v * w;
    acc += (p.x + p.y) + (p.z + p.w);
  }
  out[row] = acc;
}

// ---------------------------------------------------------------------------
extern "C" void kernel_launch(void* const* d_in, const int* in_sizes, int n_in,
                              void* d_out, int out_size, void* d_ws,
                              size_t ws_size, hipStream_t stream) {
  const float* x    = (const float*)d_in[0];
  const float* Wp   = (const float*)d_in[1];
  const float* bp   = (const float*)d_in[2];
  const float* g0   = (const float*)d_in[3];
  const float* be0  = (const float*)d_in[4];
  const float* W1s  = (const float*)d_in[5];
  const float* b1s  = (const float*)d_in[6];
  const float* g1s  = (const float*)d_in[7];
  const float* be1s = (const float*)d_in[8];
  const float* W2s  = (const float*)d_in[9];
  const float* b2s  = (const float*)d_in[10];
  const float* g2s  = (const float*)d_in[11];
  const float* be2s = (const float*)d_in[12];
  const float* Wo   = (const float*)d_in[13];
  const float* bo   = (const float*)d_in[14];

  // Workspace: H (N*32 f32) | C (N*32 f32) | partials (NGROUPS*64) | sc (33*64)
  float* Hb       = (float*)d_ws;
  float* Cb       = Hb + (size_t)NROWS * HP;
  float* partials = Cb + (size_t)NROWS * HP;
  float* sc       = partials + (size_t)NGROUPS * 64;

  proj_kernel<<<NGROUPS, TPB, 0, stream>>>(x, Wp, bp, Hb, partials);
  stats_fin_kernel<<<1, TPB, 0, stream>>>(partials, g0, be0, sc);  // BN0 -> 0

  for (int k = 0; k < 16; ++k) {
    // consumes BN affine slot 2k (BN0 for k==0, BN2 of block k-1 otherwise)
    block1_kernel<<<NGROUPS, TPB, 0, stream>>>(
        Hb, Cb, sc + (size_t)(2 * k) * 64, W1s + (size_t)k * HDIM * HDIM,
        b1s + (size_t)k * HDIM, partials, (k == 0) ? 0 : 1);
    stats_fin_kernel<<<1, TPB, 0, stream>>>(
        partials, g1s + (size_t)k * HDIM, be1s + (size_t)k * HDIM,
        sc + (size_t)(2 * k + 1) * 64);
    block2_kernel<<<NGROUPS, TPB, 0, stream>>>(
        Hb, sc + (size_t)(2 * k + 1) * 64, W2s + (size_t)k * HDIM * HDIM,
        b2s + (size_t)k * HDIM, partials);
    stats_fin_kernel<<<1, TPB, 0, stream>>>(
        partials, g2s + (size_t)k * HDIM, be2s + (size_t)k * HDIM,
        sc + (size_t)(2 * k + 2) * 64);
  }

  final_kernel<<<NROWS / TPB, TPB, 0, stream>>>(
      Hb, Cb, sc + (size_t)32 * 64, Wo, bo, (float*)d_out);
}